// ConvNetLayer_44684839748260
// MI455X (gfx1250) — compile-verified
//
#include <hip/hip_runtime.h>
#include <hip/hip_bf16.h>

#define NNODES 20000
#define NEDGES 320000
#define EK_WAVES 4

typedef __attribute__((ext_vector_type(2))) float v2f;
typedef __attribute__((ext_vector_type(8))) float v8f;
typedef __attribute__((ext_vector_type(4))) unsigned int v4u;
typedef __attribute__((ext_vector_type(4))) int v4i;
typedef __attribute__((ext_vector_type(8))) int v8i;

// D = A(16x4,f32) * B(4x16,f32) + C   -> v_wmma_f32_16x16x4_f32
__device__ __forceinline__ v8f wmma4(v2f a, v2f b, v8f c) {
  return __builtin_amdgcn_wmma_f32_16x16x4_f32(
      /*neg_a=*/false, a, /*neg_b=*/false, b,
      /*c_mod=*/(short)0, c, /*reuse_a=*/false, /*reuse_b=*/false);
}

__device__ __forceinline__ float silu(float x) { return x / (1.0f + __expf(-x)); }

// B fragment of a row-major (K x ldw) matrix; K-block k0, column block n0.
// Layout assumption (mirror of C/D striping): VGPR j holds K = k0 + j + 2*hi.
__device__ __forceinline__ v2f ldb(const float* __restrict__ W, int ldw,
                                   int k0, int n0, int col, int hi) {
  int kk = k0 + 2 * hi;
  v2f b;
  b.x = W[kk * ldw + n0 + col];
  b.y = W[(kk + 1) * ldw + n0 + col];
  return b;
}

// TDM: DMA a contiguous array of `nelem` f32 from global into LDS.
// D# per CDNA5 ISA: described as a 2D tensor with
// tensor_dim0 = tile_dim0 = nelem, tensor_dim1 = tile_dim1 = 1.
__device__ __forceinline__ void tdm_load_f32(void* lds_dst, const void* gsrc,
                                             unsigned int nelem) {
  unsigned int lds_addr = (unsigned int)(unsigned long long)lds_dst;  // low 32b = LDS offset
  unsigned long long ga = (unsigned long long)gsrc;
  v4u g0;
  g0[0] = 1u;                                            // count=1, user descriptor
  g0[1] = lds_addr;                                      // lds_addr
  g0[2] = (unsigned int)(ga & 0xFFFFFFFFu);              // global_addr[31:0]
  g0[3] = (unsigned int)((ga >> 32) & 0x1FFFFFFu)        // global_addr[56:32]
        | (2u << 30);                                    // type=2 (image)
  v8i g1;
  g1[0] = (int)(2u << 16);                               // data_size = 4 bytes
  g1[1] = (int)((nelem & 0xFFFFu) << 16);                // tensor_dim0[15:0]
  g1[2] = (int)(((nelem >> 16) & 0xFFFFu) | (1u << 16)); // tensor_dim0[31:16] | tensor_dim1[15:0]=1
  g1[3] = (int)((nelem & 0xFFFFu) << 16);                // tensor_dim1 hi=0 | tile_dim0 = nelem
  g1[4] = 1;                                             // tile_dim1 = 1, tile_dim2 = 0
  g1[5] = (int)nelem;                                    // tensor_dim0_stride[31:0]
  g1[6] = 0;
  g1[7] = 0;
  v4i z4 = {0, 0, 0, 0};
#if __clang_major__ >= 23
  v8i z8 = {0, 0, 0, 0, 0, 0, 0, 0};
  __builtin_amdgcn_tensor_load_to_lds(g0, g1, z4, z4, z8, 0);
#else
  __builtin_amdgcn_tensor_load_to_lds(g0, g1, z4, z4, 0);
#endif
}

__global__ void zero_kernel(float* __restrict__ p, long long n) {
  long long i = (long long)blockIdx.x * blockDim.x + threadIdx.x;
  if (i < n) p[i] = 0.0f;
}

// ---------------- Kernel 1: per-node linear maps y0 / y1 ----------------
// y layout (node-major, 128 floats/node): [0:32] = y0, [32 + u*3 + i] = y1[u][i]
__global__ void node_prep_kernel(const float* __restrict__ nf,
                                 const float* __restrict__ lin1_w0,
                                 const float* __restrict__ lin1_w1,
                                 float* __restrict__ y, int n) {
  int wave  = blockIdx.x * (blockDim.x >> 5) + (threadIdx.x >> 5);
  int lane  = threadIdx.x & 31;
  int tile0 = wave * 16;
  if (tile0 >= n) return;
  int row = lane & 15;
  int hi  = lane >> 4;
  int col = row;
  const float* Xr = nf + (size_t)(tile0 + row) * 128;
  const float inv_s32 = 0.17677669529663687f;  // 1/sqrt(32)

  // y0 = x0 @ lin1_w0
  v8f a0 = {}, a1 = {};
  for (int k0 = 0; k0 < 32; k0 += 4) {
    int kk = k0 + 2 * hi;
    v2f a; a.x = Xr[kk]; a.y = Xr[kk + 1];
    a0 = wmma4(a, ldb(lin1_w0, 32, k0, 0,  col, hi), a0);
    a1 = wmma4(a, ldb(lin1_w0, 32, k0, 16, col, hi), a1);
  }
  for (int r = 0; r < 8; ++r) {
    int mr = tile0 + r + 8 * hi;
    y[(size_t)mr * 128 + col]      = a0[r] * inv_s32;
    y[(size_t)mr * 128 + 16 + col] = a1[r] * inv_s32;
  }

  // y1_i = x1[:,:,i] @ lin1_w1
  for (int i = 0; i < 3; ++i) {
    v8f b0 = {}, b1 = {};
    for (int k0 = 0; k0 < 32; k0 += 4) {
      int kk = k0 + 2 * hi;
      v2f a; a.x = Xr[32 + kk * 3 + i]; a.y = Xr[32 + (kk + 1) * 3 + i];
      b0 = wmma4(a, ldb(lin1_w1, 32, k0, 0,  col, hi), b0);
      b1 = wmma4(a, ldb(lin1_w1, 32, k0, 16, col, hi), b1);
    }
    for (int r = 0; r < 8; ++r) {
      int mr = tile0 + r + 8 * hi;
      y[(size_t)mr * 128 + 32 + col * 3 + i]        = b0[r] * inv_s32;
      y[(size_t)mr * 128 + 32 + (16 + col) * 3 + i] = b1[r] * inv_s32;
    }
  }
}

// ---------------- Kernel 2: fused edge MLP + tensor product + scatter ----------------
__global__ void edge_kernel(const float* __restrict__ edge_attrs,
                            const float* __restrict__ edge_emb,
                            const float* __restrict__ fc_w1,
                            const float* __restrict__ fc_w2,
                            const float* __restrict__ fc_w3,
                            const float* __restrict__ y,
                            const int*   __restrict__ edge_index,
                            float* __restrict__ m) {
  __shared__ float w1s[8 * 64];     //  2 KB - staged via TDM
  __shared__ float w2s[64 * 64];    // 16 KB - staged via TDM
  __shared__ float w3s[64 * 128];   // 32 KB - staged via TDM
  __shared__ float h_buf[EK_WAVES][16 * 64];
  __shared__ float w_buf[EK_WAVES][16 * 128];
  int wv   = threadIdx.x >> 5;
  int lane = threadIdx.x & 31;
  int e0   = (blockIdx.x * EK_WAVES + wv) * 16;
  int row = lane & 15, hi = lane >> 4, col = row;
  float* H  = h_buf[wv];
  float* Wt = w_buf[wv];

  // Stage all MLP weights into LDS with the Tensor Data Mover (one wave issues,
  // waits on TENSORcnt, then block-wide barrier releases everyone).
  if (wv == 0) {
    tdm_load_f32(w1s, fc_w1, 8 * 64);
    tdm_load_f32(w2s, fc_w2, 64 * 64);
    tdm_load_f32(w3s, fc_w3, 64 * 128);
    __builtin_amdgcn_s_wait_tensorcnt(0);
  }
  __syncthreads();

  // layer1: silu(emb(16x8) @ fc_w1(8x64) / sqrt(8)) -> H
  {
    const float* Er = edge_emb + (size_t)(e0 + row) * 8;
    v8f acc[4] = {};
    for (int k0 = 0; k0 < 8; k0 += 4) {
      int kk = k0 + 2 * hi;
      v2f a; a.x = Er[kk]; a.y = Er[kk + 1];
      for (int nt = 0; nt < 4; ++nt)
        acc[nt] = wmma4(a, ldb(w1s, 64, k0, nt * 16, col, hi), acc[nt]);
    }
    const float s = 0.3535533905932738f;  // 1/sqrt(8)
    for (int nt = 0; nt < 4; ++nt)
      for (int r = 0; r < 8; ++r)
        H[(r + 8 * hi) * 64 + nt * 16 + col] = silu(acc[nt][r] * s);
  }
  // layer2: silu(H(16x64) @ fc_w2(64x64) / 8) -> H (in-wave LDS ops are in-order)
  {
    v8f acc[4] = {};
    for (int k0 = 0; k0 < 64; k0 += 4) {
      int kk = k0 + 2 * hi;
      v2f a; a.x = H[row * 64 + kk]; a.y = H[row * 64 + kk + 1];
      for (int nt = 0; nt < 4; ++nt)
        acc[nt] = wmma4(a, ldb(w2s, 64, k0, nt * 16, col, hi), acc[nt]);
    }
    for (int nt = 0; nt < 4; ++nt)
      for (int r = 0; r < 8; ++r)
        H[(r + 8 * hi) * 64 + nt * 16 + col] = silu(acc[nt][r] * 0.125f);
  }
  // layer3: W = H(16x64) @ fc_w3(64x128) / 8 -> Wt
  {
    v8f acc[8] = {};
    for (int k0 = 0; k0 < 64; k0 += 4) {
      int kk = k0 + 2 * hi;
      v2f a; a.x = H[row * 64 + kk]; a.y = H[row * 64 + kk + 1];
      for (int nt = 0; nt < 8; ++nt)
        acc[nt] = wmma4(a, ldb(w3s, 128, k0, nt * 16, col, hi), acc[nt]);
    }
    for (int nt = 0; nt < 8; ++nt)
      for (int r = 0; r < 8; ++r)
        Wt[(r + 8 * hi) * 128 + nt * 16 + col] = acc[nt][r] * 0.125f;
  }
  // tensor product + scatter-add (2 lanes per edge; target is L2-resident)
  {
    int el = e0 + (lane >> 1);
    int u0 = (lane & 1) * 16;
    int src = edge_index[el];
    int dst = edge_index[NEDGES + el];
    float ea0 = edge_attrs[(size_t)el * 4 + 0];
    float ex  = edge_attrs[(size_t)el * 4 + 1];
    float ey  = edge_attrs[(size_t)el * 4 + 2];
    float ez  = edge_attrs[(size_t)el * 4 + 3];
    const float* g  = y + (size_t)src * 128;
    const float* wr = Wt + (lane >> 1) * 128;
    float* md = m + (size_t)dst * 256;
    const float inv_s3 = 0.5773502691896258f;  // 1/sqrt(3)
    for (int u = u0; u < u0 + 16; ++u) {
      float w00 = wr[u], w01 = wr[32 + u], w10 = wr[64 + u], w11 = wr[96 + u];
      float g0  = g[u];
      float g1x = g[32 + u * 3], g1y = g[32 + u * 3 + 1], g1z = g[32 + u * 3 + 2];
      atomicAdd(md + u, w00 * g0 * ea0);
      float dot = g1x * ex + g1y * ey + g1z * ez;
      atomicAdd(md + 32 + u, w11 * dot * inv_s3);
      float a01 = w01 * g0;
      atomicAdd(md + 64 + u * 3 + 0, a01 * ex);
      atomicAdd(md + 64 + u * 3 + 1, a01 * ey);
      atomicAdd(md + 64 + u * 3 + 2, a01 * ez);
      float a10 = w10 * ea0;
      atomicAdd(md + 160 + u * 3 + 0, a10 * g1x);
      atomicAdd(md + 160 + u * 3 + 1, a10 * g1y);
      atomicAdd(md + 160 + u * 3 + 2, a10 * g1z);
    }
  }
}

// ---------------- Kernel 3: node finalize (sc einsums + lin2 + gate + residual) ----------------
__global__ void node_final_kernel(const float* __restrict__ nf,
                                  const float* __restrict__ attrs,
                                  const float* __restrict__ m,
                                  const float* __restrict__ lin2_w0,
                                  const float* __restrict__ lin2_w1,
                                  const float* __restrict__ sc_w0,
                                  const float* __restrict__ sc_w1,
                                  const int*   __restrict__ avg,
                                  float* __restrict__ out, int n) {
  int wave  = blockIdx.x * (blockDim.x >> 5) + (threadIdx.x >> 5);
  int lane  = threadIdx.x & 31;
  int tile0 = wave * 16;
  if (tile0 >= n) return;
  int row = lane & 15, hi = lane >> 4, col = row;
  int r = tile0 + row;
  const float* Xr = nf    + (size_t)r * 128;
  const float* Ar = attrs + (size_t)r * 4;
  const float* Mr = m     + (size_t)r * 256;
  float s_m = rsqrtf((float)avg[0]) * 0.125f;       // 1/sqrt(avg) * 1/sqrt(64)
  const float sc_norm = 0.08838834764831845f;       // 1/sqrt(32*4)

  // out0 (16x64): sc0 + s0 @ lin2_w0 / 8
  v8f o0[4] = {};
  for (int k0 = 0; k0 < 128; k0 += 4) {            // A = flat outer(x0, attrs)
    int kk = k0 + 2 * hi;
    v2f a;
    a.x = Xr[kk >> 2] * Ar[kk & 3];
    a.y = Xr[(kk + 1) >> 2] * Ar[(kk + 1) & 3];
    for (int nt = 0; nt < 4; ++nt)
      o0[nt] = wmma4(a, ldb(sc_w0, 64, k0, nt * 16, col, hi), o0[nt]);
  }
  for (int nt = 0; nt < 4; ++nt)
    for (int rr = 0; rr < 8; ++rr) o0[nt][rr] *= sc_norm;
  for (int k0 = 0; k0 < 64; k0 += 4) {             // A = m[:, :64] * s_m
    int kk = k0 + 2 * hi;
    v2f a; a.x = Mr[kk] * s_m; a.y = Mr[kk + 1] * s_m;
    for (int nt = 0; nt < 4; ++nt)
      o0[nt] = wmma4(a, ldb(lin2_w0, 64, k0, nt * 16, col, hi), o0[nt]);
  }

  // out1 (16x32 x3): sc1_i + s1_i @ lin2_w1 / 8
  v8f o1[3][2] = {};
  for (int i = 0; i < 3; ++i) {
    for (int k0 = 0; k0 < 128; k0 += 4) {          // A = flat outer(x1_i, attrs)
      int kk = k0 + 2 * hi;
      v2f a;
      a.x = Xr[32 + (kk >> 2) * 3 + i] * Ar[kk & 3];
      a.y = Xr[32 + ((kk + 1) >> 2) * 3 + i] * Ar[(kk + 1) & 3];
      for (int nt = 0; nt < 2; ++nt)
        o1[i][nt] = wmma4(a, ldb(sc_w1, 32, k0, nt * 16, col, hi), o1[i][nt]);
    }
    for (int nt = 0; nt < 2; ++nt)
      for (int rr = 0; rr < 8; ++rr) o1[i][nt][rr] *= sc_norm;
    for (int k0 = 0; k0 < 64; k0 += 4) {           // A = m[:, 64 + k*3 + i] * s_m
      int kk = k0 + 2 * hi;
      v2f a; a.x = Mr[64 + kk * 3 + i] * s_m; a.y = Mr[64 + (kk + 1) * 3 + i] * s_m;
      for (int nt = 0; nt < 2; ++nt)
        o1[i][nt] = wmma4(a, ldb(lin2_w1, 32, k0, nt * 16, col, hi), o1[i][nt]);
    }
  }

  // gate + residual
  for (int rr = 0; rr < 8; ++rr) {
    int mr = tile0 + rr + 8 * hi;
    float* Or = out + (size_t)mr * 128;
    const float* NR = nf + (size_t)mr * 128;
    Or[col]      = NR[col]      + silu(o0[0][rr]);
    Or[16 + col] = NR[16 + col] + silu(o0[1][rr]);
    float gA = silu(o0[2][rr]);
    float gB = silu(o0[3][rr]);
    for (int i = 0; i < 3; ++i) {
      Or[32 + col * 3 + i]        = NR[32 + col * 3 + i]        + gA * o1[i][0][rr];
      Or[32 + (16 + col) * 3 + i] = NR[32 + (16 + col) * 3 + i] + gB * o1[i][1][rr];
    }
  }
}

extern "C" void kernel_launch(void* const* d_in, const int* in_sizes, int n_in,
                              void* d_out, int out_size, void* d_ws, size_t ws_size,
                              hipStream_t stream) {
  const float* nf    = (const float*)d_in[0];
  const float* attrs = (const float*)d_in[1];
  const float* eatt  = (const float*)d_in[2];
  const float* eemb  = (const float*)d_in[3];
  const float* l1w0  = (const float*)d_in[4];
  const float* l1w1  = (const float*)d_in[5];
  const float* fcw1  = (const float*)d_in[6];
  const float* fcw2  = (const float*)d_in[7];
  const float* fcw3  = (const float*)d_in[8];
  const float* l2w0  = (const float*)d_in[9];
  const float* l2w1  = (const float*)d_in[10];
  const float* scw0  = (const float*)d_in[11];
  const float* scw1  = (const float*)d_in[12];
  const int*   eidx  = (const int*)d_in[13];
  const int*   avg   = (const int*)d_in[14];
  float* out = (float*)d_out;

  float* y = (float*)d_ws;                         // N*128 floats (10.24 MB)
  float* m = (float*)d_ws + (size_t)NNODES * 128;  // N*256 floats (20.48 MB)

  long long mElems = (long long)NNODES * 256;
  zero_kernel<<<(int)((mElems + 255) / 256), 256, 0, stream>>>(m, mElems);

  int nodeTiles = NNODES / 16;  // 1250
  node_prep_kernel<<<(nodeTiles + 7) / 8, 256, 0, stream>>>(nf, l1w0, l1w1, y, NNODES);

  edge_kernel<<<NEDGES / (16 * EK_WAVES), 32 * EK_WAVES, 0, stream>>>(
      eatt, eemb, fcw1, fcw2, fcw3, y, eidx, m);

  node_final_kernel<<<(nodeTiles + 3) / 4, 128, 0, stream>>>(
      nf, attrs, m, l2w0, l2w1, scw0, scw1, avg, out, NNODES);
}